// WeightsDropout_10582799418055
// MI455X (gfx1250) — compile-verified
//
#include <hip/hip_runtime.h>
#include <cstdint>

// WeightsDropout for w[8192, 1, 4096] fp32:
//   per row: zero the k = N/2 smallest entries (stable-sort tie order),
//   softmax the survivors, dropped entries stay 0.
//
// Memory-bound (256 MB @ 23.3 TB/s ~ 11 us). One workgroup (8 wave32) per row.
// Row staged in LDS via CDNA5 async global->LDS DMA (ASYNCcnt). Exact k-th
// smallest via 4-pass radix select on fp32 bit patterns (positive floats are
// order-preserving as uints) with per-wave bank-staggered histograms and a
// wave-cooperative shfl-scan bin select. All reductions are deterministic.

#define NCOLS 4096
#define KDROP 2048           // int(N * P), P = 0.5
#define NTHREADS 256
#define NWAVES 8             // 256 / wave32
#define HSTRIDE 264          // 256 bins + 8 pad dwords -> copy c shifts banks by 8c
#define EPT 16               // elements per thread

typedef float v4f __attribute__((ext_vector_type(4)));

__global__ __launch_bounds__(NTHREADS) void weights_dropout_kernel(
    const float* __restrict__ w, float* __restrict__ out) {
  __shared__ uint32_t s_bits[NCOLS];              // 16 KB: row as raw bits
  __shared__ uint32_t s_hist[NWAVES * HSTRIDE];   // per-wave histogram copies
  __shared__ uint32_t s_comb[256];                // combined histogram
  __shared__ float    s_red[NWAVES];              // per-wave partial sums
  __shared__ uint32_t s_prefix;                   // radix prefix -> k-th value bits
  __shared__ uint32_t s_rank;                     // residual rank in candidates
  __shared__ uint32_t s_maxbits;                  // row max (uint cmp == fp cmp, all > 0)
  __shared__ float    s_total;                    // partition function

  const int tid  = threadIdx.x;
  const int lane = tid & 31;
  const int wave = tid >> 5;
  const long long row = blockIdx.x;
  const uint64_t gsrc = (uint64_t)(uintptr_t)(w + row * (long long)NCOLS);

  // ---- CDNA5 async global -> LDS row copy (GLOBAL_LOAD_ASYNC_TO_LDS_B128) ----
  // 256 lanes x 16 B = 4 KB per issue; 4 issues cover the 16 KB row.
  const uint32_t ldsBase = (uint32_t)(uintptr_t)(&s_bits[0]);
#pragma unroll
  for (int c = 0; c < 4; ++c) {
    const uint32_t off = (uint32_t)(c * 4096 + tid * 16);  // byte offset
    const uint32_t ldsAddr = ldsBase + off;
    asm volatile("global_load_async_to_lds_b128 %0, %1, %2 offset:0"
                 :
                 : "v"(ldsAddr), "v"(off), "s"(gsrc)
                 : "memory");
  }
  asm volatile("s_wait_asynccnt 0" ::: "memory");  // this wave's DMA done

  if (tid == 0) {
    s_prefix  = 0u;
    s_rank    = (uint32_t)KDROP;  // target sorted index (0-based) = k
    s_maxbits = 0u;
  }
  __syncthreads();  // all waves' LDS data + init visible

  // ---- pull this thread's 16 elements into registers (b128 LDS loads) ----
  uint32_t myb[EPT];
#pragma unroll
  for (int c = 0; c < 4; ++c) {
    uint4 q = *reinterpret_cast<const uint4*>(&s_bits[c * 1024 + tid * 4]);
    myb[c * 4 + 0] = q.x;
    myb[c * 4 + 1] = q.y;
    myb[c * 4 + 2] = q.z;
    myb[c * 4 + 3] = q.w;
  }

  // ---- row max: butterfly reduce per wave, one atomic per wave ----
  uint32_t lmax = 0u;
#pragma unroll
  for (int e = 0; e < EPT; ++e) lmax = (myb[e] > lmax) ? myb[e] : lmax;
#pragma unroll
  for (int off = 16; off > 0; off >>= 1) {
    const uint32_t o = __shfl_xor(lmax, off);
    lmax = (o > lmax) ? o : lmax;
  }
  if (lane == 0) atomicMax(&s_maxbits, lmax);

  // ---- 4-pass radix select: value with sorted (ascending) index KDROP ----
  for (int shift = 24; shift >= 0; shift -= 8) {
    // zero this pass's bins in every wave copy
#pragma unroll
    for (int c = 0; c < NWAVES; ++c) s_hist[c * HSTRIDE + tid] = 0u;
    __syncthreads();

    const uint32_t prefix = s_prefix;
    const uint32_t mask = (shift == 24) ? 0u : (0xFFFFFFFFu << (shift + 8));
    uint32_t* const whist = &s_hist[wave * HSTRIDE];
#pragma unroll
    for (int e = 0; e < EPT; ++e) {
      const uint32_t b = myb[e];
      if ((b & mask) == prefix) atomicAdd(&whist[(b >> shift) & 255u], 1u);
    }
    __syncthreads();

    // combine the wave copies (conflict-free: stride 264 shifts banks by 8)
    uint32_t tot = 0u;
#pragma unroll
    for (int c = 0; c < NWAVES; ++c) tot += s_hist[c * HSTRIDE + tid];
    s_comb[tid] = tot;
    __syncthreads();

    // wave 0 cooperative bin select: lane owns 8 bins, shfl_up exclusive scan
    if (tid < 32) {
      uint32_t h[8];
      uint32_t lsumb = 0u;
#pragma unroll
      for (int j = 0; j < 8; ++j) {
        h[j] = s_comb[tid * 8 + j];
        lsumb += h[j];
      }
      uint32_t inc = lsumb;
#pragma unroll
      for (int off = 1; off < 32; off <<= 1) {
        const uint32_t n = __shfl_up(inc, off);
        if (tid >= off) inc += n;
      }
      const uint32_t exc = inc - lsumb;
      const uint32_t r = s_rank;
      if (r >= exc && r < exc + lsumb) {  // exactly one lane
        uint32_t rr = r - exc;
        uint32_t dsel = 0u;
        bool found = false;
#pragma unroll
        for (int j = 0; j < 8; ++j) {
          if (!found) {
            if (rr < h[j]) {
              dsel = (uint32_t)(tid * 8 + j);
              found = true;
            } else {
              rr -= h[j];
            }
          }
        }
        s_rank   = rr;                       // residual rank inside chosen bin
        s_prefix = prefix | (dsel << shift);
      }
    }
    __syncthreads();
  }

  const uint32_t tbits = s_prefix;  // bits of k-th smallest value
  const uint32_t req   = s_rank;    // # of ties (by ascending index) dropped
  const float    m     = __uint_as_float(s_maxbits);

  // ---- keep mask + exp; ties follow stable-argsort (index-ascending) order --
  float ex[EPT];
  float lsum = 0.0f;
#pragma unroll
  for (int c = 0; c < 4; ++c) {
#pragma unroll
    for (int j = 0; j < 4; ++j) {
      const int e   = c * 4 + j;
      const int idx = c * 1024 + tid * 4 + j;
      const uint32_t b = myb[e];
      bool kp;
      if (b != tbits) {
        kp = (b > tbits);
      } else {
        // rare tie path: equals with eq_index < req are dropped
        uint32_t eqi = 0u;
        for (int q = 0; q < idx; ++q) eqi += (s_bits[q] == tbits) ? 1u : 0u;
        kp = (eqi >= req);
      }
      const float v = kp ? expf(__uint_as_float(b) - m) : 0.0f;
      ex[e] = v;
      lsum += v;
    }
  }

  // ---- deterministic partition function: wave butterfly + serial 8-way ----
#pragma unroll
  for (int off = 16; off > 0; off >>= 1) lsum += __shfl_xor(lsum, off);
  if (lane == 0) s_red[wave] = lsum;
  __syncthreads();
  if (tid == 0) {
    float t = 0.0f;
#pragma unroll
    for (int i = 0; i < NWAVES; ++i) t += s_red[i];
    s_total = t;
  }
  __syncthreads();
  const float total = s_total;

  // ---- normalized output, nontemporal b128 stores (write-once stream) ----
  float* gdst = out + row * (long long)NCOLS;
#pragma unroll
  for (int c = 0; c < 4; ++c) {
    v4f o;
    o.x = ex[c * 4 + 0] / total;
    o.y = ex[c * 4 + 1] / total;
    o.z = ex[c * 4 + 2] / total;
    o.w = ex[c * 4 + 3] / total;
    __builtin_nontemporal_store(o, reinterpret_cast<v4f*>(&gdst[c * 1024 + tid * 4]));
  }
}

extern "C" void kernel_launch(void* const* d_in, const int* in_sizes, int n_in,
                              void* d_out, int out_size, void* d_ws, size_t ws_size,
                              hipStream_t stream) {
  const float* w = (const float*)d_in[0];
  float* out = (float*)d_out;
  const int rows = in_sizes[0] / NCOLS;  // 8192 * 1
  weights_dropout_kernel<<<dim3(rows), dim3(NTHREADS), 0, stream>>>(w, out);
}